// DigitCap_54365696033056
// MI455X (gfx1250) — compile-verified
//
#include <hip/hip_runtime.h>
#include <math.h>

// ---- problem constants ----
#define BB   256
#define RR   1152
#define CC   10
#define DI   8
#define DO   16
#define NCOL (CC*DO)      // 160
#define KK   (RR*DI)      // 9216
#define EPS  1e-5f
#define SWAVES 8          // K-split waves in s_kernel (RR % SWAVES == 0)

typedef __attribute__((ext_vector_type(2))) float v2f;
typedef __attribute__((ext_vector_type(8))) float v8f;

__device__ inline v2f ld2(const float* p) { return *reinterpret_cast<const v2f*>(p); }

__device__ inline float squashf(float s) {
    float sq = s * s;
    return sq / (1.0f + sq) * s / (sqrtf(sq) + EPS);
}

// ---------------------------------------------------------------------------
// K0: Wsum[r,c,i] = (1/B) * sum_d W[r,c,d,i]
// ---------------------------------------------------------------------------
__global__ void wsum_kernel(const float* __restrict__ W, float* __restrict__ Wsum) {
    int idx = blockIdx.x * blockDim.x + threadIdx.x;   // over R*C*8 = 92160
    if (idx >= RR * CC * DI) return;
    int i  = idx & 7;
    int rc = idx >> 3;
    const float* w = W + rc * (DO * DI) + i;
    float s = 0.0f;
#pragma unroll
    for (int d = 0; d < DO; ++d) s += w[d * DI];
    Wsum[idx] = s * (1.0f / (float)BB);
}

// ---------------------------------------------------------------------------
// K1: t[r,c,b] = sum_i Wsum[r,c,i] * u[b,r,i]     (1/B already folded in Wsum)
// grid = R*C blocks, block = 256 threads (b)
// ---------------------------------------------------------------------------
__global__ void t_kernel(const float* __restrict__ u, const float* __restrict__ Wsum,
                         float* __restrict__ t) {
    int rc = blockIdx.x;           // r*10 + c
    int r  = rc / CC;
    int b  = threadIdx.x;
    const float* up = u + (size_t)b * KK + r * DI;
    const float* wp = Wsum + rc * DI;
    float s = 0.0f;
#pragma unroll
    for (int i = 0; i < DI; ++i) s += up[i] * wp[i];
    t[(size_t)rc * BB + b] = s;
}

// ---------------------------------------------------------------------------
// K2: s/v kernel (main WMMA GEMM, fp32 16x16x4):
//   s[b,c,d] = sum_{r,i} u[b,r,i] * scale(r,c,d) * W[r,c,d,i];  v = squash(s)
// USE_CIJ: scale = cij[r*160 + c*16 + d], else uniform 1/R.
// FINAL_OUT: write (b, c*16+d) layout (d_out); else transposed vT[(c*16+d)*256+b]
// grid = (16 batch tiles, 10 capsules), block = 32*SWAVES (K split over r)
// ---------------------------------------------------------------------------
template <bool USE_CIJ, bool FINAL_OUT>
__global__ void s_kernel(const float* __restrict__ u, const float* __restrict__ W,
                         const float* __restrict__ cij,
                         float* __restrict__ vout) {
    const int mt   = blockIdx.x;            // batch tile (16 rows)
    const int c    = blockIdx.y;            // capsule
    const int wv   = threadIdx.x >> 5;      // 0..SWAVES-1 (K-split)
    const int lane = threadIdx.x & 31;
    const int lm   = lane & 15;             // row m (A) / col d (B, C/D)
    const int lk   = lane >> 4;             // K half select

    const float* uA = u + (size_t)(mt * 16 + lm) * KK;   // A row base
    const float* Wc = W + c * (DO * DI);                 // stride CC*DO*DI per r
    const float invR = 1.0f / (float)RR;

    v8f acc0 = {}; v8f acc1 = {};

    const int r0 = wv * (RR / SWAVES);
    const int r1 = r0 + (RR / SWAVES);
    for (int r = r0; r < r1; ++r) {
        const float* wr = Wc + (size_t)r * (CC * DO * DI);
        float scale = USE_CIJ ? cij[r * NCOL + c * DO + lm] : invR;  // per column d=lm
        // A tile: element (m=lm, k=lk*2+j), i-offset 0 / 4
        v2f a0 = ld2(uA + r * DI + lk * 2);
        v2f a1 = ld2(uA + r * DI + 4 + lk * 2);
        // B tile: element (k=lk*2+j, n=d=lm) = W[r,c,d,i] * scale
        v2f b0 = ld2(wr + lm * DI + lk * 2) * scale;
        v2f b1 = ld2(wr + lm * DI + 4 + lk * 2) * scale;
        acc0 = __builtin_amdgcn_wmma_f32_16x16x4_f32(false, a0, false, b0,
                                                     (short)0, acc0, false, false);
        acc1 = __builtin_amdgcn_wmma_f32_16x16x4_f32(false, a1, false, b1,
                                                     (short)0, acc1, false, false);
    }
    v8f acc = acc0 + acc1;

    __shared__ float sred[SWAVES][32][8];
#pragma unroll
    for (int g = 0; g < 8; ++g) sred[wv][lane][g] = acc[g];
    __syncthreads();

    if (wv == 0) {
#pragma unroll
        for (int g = 0; g < 8; ++g) {
            float s = 0.0f;
#pragma unroll
            for (int w = 0; w < SWAVES; ++w) s += sred[w][lane][g];
            float v = squashf(s);
            int b = mt * 16 + g + 8 * lk;   // C/D layout: VGPR g, lane half -> M
            if (FINAL_OUT)
                vout[(size_t)b * NCOL + c * DO + lm] = v;            // (b,c,d)
            else
                vout[(size_t)(c * DO + lm) * BB + b] = v;            // vT[c,d,b]
        }
    }
}

// ---------------------------------------------------------------------------
// K3: routing-logit update (WMMA GEMM per capsule):
//   bnew[r,c,d] = (first ? 0 : bold[r,c,d]) + sum_b t[r,c,b] * vT[c,d,b]
// grid = (72 r-tiles, 10 capsules), block = 32 (1 wave)
// ---------------------------------------------------------------------------
template <bool FIRST>
__global__ void upd_kernel(const float* __restrict__ t, const float* __restrict__ vT,
                           const float* __restrict__ bold, float* __restrict__ bnew) {
    const int rt   = blockIdx.x;            // r tile (16 rows)
    const int c    = blockIdx.y;
    const int lane = threadIdx.x & 31;
    const int lm   = lane & 15;
    const int lk   = lane >> 4;

    const float* tA = t + ((size_t)(rt * 16 + lm) * CC + c) * BB;   // A row (k=b)
    const float* vB = vT + (size_t)(c * DO + lm) * BB;              // B row (k=b), d=lm

    v8f acc0 = {}; v8f acc1 = {};
    for (int kb = 0; kb < BB; kb += 8) {
        v2f a0 = ld2(tA + kb + lk * 2);
        v2f a1 = ld2(tA + kb + 4 + lk * 2);
        v2f b0 = ld2(vB + kb + lk * 2);
        v2f b1 = ld2(vB + kb + 4 + lk * 2);
        acc0 = __builtin_amdgcn_wmma_f32_16x16x4_f32(false, a0, false, b0,
                                                     (short)0, acc0, false, false);
        acc1 = __builtin_amdgcn_wmma_f32_16x16x4_f32(false, a1, false, b1,
                                                     (short)0, acc1, false, false);
    }
#pragma unroll
    for (int g = 0; g < 8; ++g) {
        int r   = rt * 16 + g + 8 * lk;
        int idx = r * NCOL + c * DO + lm;
        float s = acc0[g] + acc1[g];
        bnew[idx] = FIRST ? s : (bold[idx] + s);
    }
}

// ---------------------------------------------------------------------------
// K4: softmax over r (axis=1 of the reference) per (c,d) column
// grid = 160 blocks (one per c*16+d), block = 256
// ---------------------------------------------------------------------------
__global__ void softmax_kernel(const float* __restrict__ b, float* __restrict__ cij) {
    const int cd  = blockIdx.x;       // c*16 + d
    const int tid = threadIdx.x;
    __shared__ float red[256];

    float mx = -3.0e38f;
    for (int r = tid; r < RR; r += 256) mx = fmaxf(mx, b[r * NCOL + cd]);
    red[tid] = mx; __syncthreads();
    for (int s = 128; s > 0; s >>= 1) {
        if (tid < s) red[tid] = fmaxf(red[tid], red[tid + s]);
        __syncthreads();
    }
    mx = red[0]; __syncthreads();

    float sum = 0.0f;
    for (int r = tid; r < RR; r += 256) sum += expf(b[r * NCOL + cd] - mx);
    red[tid] = sum; __syncthreads();
    for (int s = 128; s > 0; s >>= 1) {
        if (tid < s) red[tid] += red[tid + s];
        __syncthreads();
    }
    float inv = 1.0f / red[0];

    for (int r = tid; r < RR; r += 256)
        cij[r * NCOL + cd] = expf(b[r * NCOL + cd] - mx) * inv;
}

// ---------------------------------------------------------------------------
extern "C" void kernel_launch(void* const* d_in, const int* in_sizes, int n_in,
                              void* d_out, int out_size, void* d_ws, size_t ws_size,
                              hipStream_t stream) {
    const float* u = (const float*)d_in[0];   // (256,1152,8)
    const float* W = (const float*)d_in[1];   // (1152,10,16,8)
    float* out = (float*)d_out;               // (256,10,16)

    float* ws   = (float*)d_ws;
    float* Wsum = ws;                          //  92160 f
    float* t    = Wsum + RR * CC * DI;         // 2949120 f (11.8 MB)
    float* vT   = t + (size_t)RR * CC * BB;    //  40960 f   (layout [c*16+d][b])
    float* blog = vT + NCOL * BB;              // 184320 f
    float* cij  = blog + RR * NCOL;            // 184320 f  (total ~13.8 MB)

    wsum_kernel<<<(RR * CC * DI + 255) / 256, 256, 0, stream>>>(W, Wsum);
    t_kernel<<<RR * CC, 256, 0, stream>>>(u, Wsum, t);

    // iteration 0: uniform routing weights (softmax of zeros over R)
    s_kernel<false, false><<<dim3(16, CC), 32 * SWAVES, 0, stream>>>(u, W, nullptr, vT);
    upd_kernel<true><<<dim3(RR / 16, CC), 32, 0, stream>>>(t, vT, blog, blog);
    softmax_kernel<<<NCOL, 256, 0, stream>>>(blog, cij);

    // iteration 1
    s_kernel<true, false><<<dim3(16, CC), 32 * SWAVES, 0, stream>>>(u, W, cij, vT);
    upd_kernel<false><<<dim3(RR / 16, CC), 32, 0, stream>>>(t, vT, blog, blog);
    softmax_kernel<<<NCOL, 256, 0, stream>>>(blog, cij);

    // iteration 2: final, squash straight into d_out
    s_kernel<true, true><<<dim3(16, CC), 32 * SWAVES, 0, stream>>>(u, W, cij, out);
}